// CompressAttn_88235808129250
// MI455X (gfx1250) — compile-verified
//
#include <hip/hip_runtime.h>
#include <math.h>

typedef __attribute__((ext_vector_type(2))) float v2f;
typedef __attribute__((ext_vector_type(8))) float v8f;

namespace {
constexpr int kB   = 2;
constexpr int kM   = 4096;
constexpr int kQH  = 16;
constexpr int kD   = 192;
constexpr int kN   = 255;
constexpr int kVD  = 128;
constexpr int kKS  = 32;    // kernel_size
constexpr int kSTR = 16;    // stride
constexpr int kNT  = 16;    // padded column tiles (256 cols)
constexpr int kK1  = kD / 4;        // 48 k-steps for Q*K^T
constexpr int kK3  = (kNT * 16) / 4; // 64 k-steps for P*V
constexpr int kPL  = 260;   // padded LDS row stride in floats (bank-conflict free)
constexpr float kScale = 0.07216878364870322993f; // 192^-0.5
}

__global__ __launch_bounds__(32)
void compress_attn_f32wmma(const float* __restrict__ q,
                           const float* __restrict__ kmat,
                           const float* __restrict__ vmat,
                           float* __restrict__ o_out,
                           float* __restrict__ p_out)
{
    __shared__ float p_lds[16 * kPL];

    const int lane = threadIdx.x & 31;
    const int half = lane >> 4;   // 0: lanes 0-15, 1: lanes 16-31
    const int l15  = lane & 15;

    const int mt = blockIdx.x & (kM / 16 - 1);          // 0..255
    const int h  = (blockIdx.x >> 8) & (kQH - 1);       // 0..15
    const int b  =  blockIdx.x >> 12;                   // 0..1
    const int m0 = mt * 16;

    const float* qrow = q    + (((size_t)b * kM + (m0 + l15)) * kQH + h) * kD;
    const float* kbh  = kmat + ((size_t)b * kN * kQH + h) * kD;
    const float* vbh  = vmat + ((size_t)b * kN * kQH + h) * kVD;

    const v8f vzero = {};

    // ---------------- Phase 1: S = Q * K^T ----------------
    v8f acc[kNT];
#pragma unroll
    for (int t = 0; t < kNT; ++t) acc[t] = vzero;

#pragma unroll 2
    for (int kk = 0; kk < kK1; ++kk) {
        const int d = kk * 4 + half * 2;               // A layout: lane half picks K pair
        const v2f a = *(const v2f*)(qrow + d);
#pragma unroll
        for (int t = 0; t < kNT; ++t) {
            int n   = t * 16 + l15;
            int nld = (n < kN) ? n : (kN - 1);         // clamp pad col; masked to -inf later
            const v2f bf = *(const v2f*)(kbh + (size_t)nld * (kQH * kD) + d);
            acc[t] = __builtin_amdgcn_wmma_f32_16x16x4_f32(
                false, a, false, bf, (short)0, acc[t], false, false);
        }
    }

    // ---------------- Phase 2: mask + scale + softmax ----------------
    // C/D layout: reg i, lane half -> row = m0 + i + 8*half ; col = t*16 + l15
#pragma unroll
    for (int t = 0; t < kNT; ++t) {
        const int n   = t * 16 + l15;
        const int thr = n * kSTR + (kKS - 1);          // masked if row < thr
#pragma unroll
        for (int i = 0; i < 8; ++i) {
            const int row = m0 + i + 8 * half;
            float s = acc[t][i] * kScale;
            s = (row < thr) ? -3.402823466e38f : s;    // finfo(f32).min like reference
            s = (n >= kN) ? -__builtin_inff() : s;     // pad column -> exp() == 0
            acc[t][i] = s;
        }
    }

#pragma unroll
    for (int i = 0; i < 8; ++i) {
        float m = -__builtin_inff();
#pragma unroll
        for (int t = 0; t < kNT; ++t) m = fmaxf(m, acc[t][i]);
#pragma unroll
        for (int off = 1; off < 16; off <<= 1)         // reduce within each 16-lane half
            m = fmaxf(m, __shfl_xor(m, off, 32));

        float ssum = 0.0f;
#pragma unroll
        for (int t = 0; t < kNT; ++t) {
            float e = __expf(acc[t][i] - m);
            acc[t][i] = e;
            ssum += e;
        }
#pragma unroll
        for (int off = 1; off < 16; off <<= 1)
            ssum += __shfl_xor(ssum, off, 32);

        const float rinv = 1.0f / ssum;
#pragma unroll
        for (int t = 0; t < kNT; ++t) acc[t][i] *= rinv;
    }

    // ---------------- write P (global, nontemporal) + stage to LDS ----------------
    float* pbh = p_out + ((size_t)b * kQH + h) * (size_t)kM * kN;
#pragma unroll
    for (int t = 0; t < kNT; ++t) {
        const int n = t * 16 + l15;
#pragma unroll
        for (int i = 0; i < 8; ++i) {
            const int rl = i + 8 * half;
            const float pv = acc[t][i];
            p_lds[rl * kPL + n] = pv;
            if (n < kN)
                __builtin_nontemporal_store(pv, pbh + (size_t)(m0 + rl) * kN + n);
        }
    }
    __syncthreads();

    // ---------------- Phase 3: O = P * V ----------------
    v8f oacc[8];
#pragma unroll
    for (int vt = 0; vt < 8; ++vt) oacc[vt] = vzero;

#pragma unroll 2
    for (int kt = 0; kt < kK3; ++kt) {
        const int kb = kt * 4 + half * 2;
        // A fragment from LDS: P[row = l15][k = kb .. kb+1] (8B aligned)
        const v2f a = *(const v2f*)(&p_lds[l15 * kPL + kb]);
        const int n0 = (kb     < kN) ? kb     : (kN - 1);  // pad rows: P col == 0 exactly
        const int n1 = (kb + 1 < kN) ? kb + 1 : (kN - 1);
        const float* v0 = vbh + (size_t)n0 * (kQH * kVD);
        const float* v1 = vbh + (size_t)n1 * (kQH * kVD);
#pragma unroll
        for (int vt = 0; vt < 8; ++vt) {
            const int vd = vt * 16 + l15;
            v2f bf;
            bf.x = v0[vd];
            bf.y = v1[vd];
            oacc[vt] = __builtin_amdgcn_wmma_f32_16x16x4_f32(
                false, a, false, bf, (short)0, oacc[vt], false, false);
        }
    }

    // ---------------- store O ----------------
#pragma unroll
    for (int vt = 0; vt < 8; ++vt) {
#pragma unroll
        for (int i = 0; i < 8; ++i) {
            const int row = m0 + i + 8 * half;
            float val = oacc[vt][i];
            if (row < kKS - 1) val = 0.0f;             // queries before first full block
            __builtin_nontemporal_store(
                val, o_out + (((size_t)b * kM + row) * kQH + h) * kVD + vt * 16 + l15);
        }
    }
}

extern "C" void kernel_launch(void* const* d_in, const int* in_sizes, int n_in,
                              void* d_out, int out_size, void* d_ws, size_t ws_size,
                              hipStream_t stream) {
    (void)in_sizes; (void)n_in; (void)d_ws; (void)ws_size; (void)out_size;
    const float* q = (const float*)d_in[0];
    const float* k = (const float*)d_in[1];
    const float* v = (const float*)d_in[2];
    // d_in[3..5] = real_length, kernel_size, stride (fixed by the reference config)

    float* o_out = (float*)d_out;                                  // [B,M,QH,VD]
    float* p_out = o_out + (size_t)kB * kM * kQH * kVD;            // [B,QH,M,N]

    const int nblocks = kB * kQH * (kM / 16);                      // 8192
    compress_attn_f32wmma<<<dim3(nblocks), dim3(32), 0, stream>>>(q, k, v, o_out, p_out);
}